// GCN_65206193488149
// MI455X (gfx1250) — compile-verified
//
#include <hip/hip_runtime.h>

typedef float v2f __attribute__((ext_vector_type(2)));
typedef float v8f __attribute__((ext_vector_type(8)));

#define NNODES 100000
#define NEDGES 800000
#define INDIM  128
#define HID1   64
#define HID2   64
#define ODIM   32

// ---------------------------------------------------------------------------
// utility: zero a float buffer
// ---------------------------------------------------------------------------
__global__ __launch_bounds__(256) void zero_f32_k(float* __restrict__ p, int n) {
    int i = blockIdx.x * blockDim.x + threadIdx.x;
    if (i < n) p[i] = 0.0f;
}

// ---------------------------------------------------------------------------
// degree: deg[dst[e]] += 1  (atomic fp32 add -> global_atomic_add_f32)
// ---------------------------------------------------------------------------
__global__ __launch_bounds__(256) void degree_k(const int* __restrict__ dst,
                                                float* __restrict__ deg, int nE) {
    int e = blockIdx.x * blockDim.x + threadIdx.x;
    if (e < nE) unsafeAtomicAdd(&deg[dst[e]], 1.0f);
}

// dis[i] = rsqrt(deg[i] + 1)   (self-loop)
__global__ __launch_bounds__(256) void rsqrt_k(float* __restrict__ d, int n) {
    int i = blockIdx.x * blockDim.x + threadIdx.x;
    if (i < n) d[i] = rsqrtf(d[i] + 1.0f);
}

// ---------------------------------------------------------------------------
// WMMA fp32 GEMM:  C[M,N] = A[M,K] * W[K,N]
// One wave32 per 16-row stripe, N/16 accumulators, K stepped by 4 with
// V_WMMA_F32_16X16X4_F32.
//   A tile 16x4 (f32, 2 VGPR/lane): lanes 0-15 -> K={k0,k0+1}, lanes 16-31 -> K={k0+2,k0+3}
//   B tile 4x16 mirrored: vgpr j, lane-half kh -> element (k0+2*kh+j, n0+lane&15)
//   C/D: vgpr v, lane-half kh -> row m0+8*kh+v, col n0+lane&15
// ---------------------------------------------------------------------------
template <int N>
__global__ __launch_bounds__(256) void gemm_wmma_f32_k(const float* __restrict__ A,
                                                       const float* __restrict__ W,
                                                       float* __restrict__ C,
                                                       int M, int K) {
    constexpr int NT = N / 16;
    const int lane = threadIdx.x & 31;
    const int wave = (blockIdx.x * blockDim.x + threadIdx.x) >> 5;
    const int m0   = wave << 4;
    if (m0 >= M) return;

    const int mr = lane & 15;   // row (A) / col (B,C) within tile
    const int kh = lane >> 4;   // K-half selector

    v8f acc[NT];
#pragma unroll
    for (int t = 0; t < NT; ++t) acc[t] = {};

    const size_t arow = (size_t)(m0 + mr) * (size_t)K;

    for (int k0 = 0; k0 < K; k0 += 4) {
        v2f a;
        a.x = A[arow + k0 + 2 * kh + 0];
        a.y = A[arow + k0 + 2 * kh + 1];
#pragma unroll
        for (int t = 0; t < NT; ++t) {
            const int ncol = (t << 4) + mr;
            v2f b;
            b.x = W[(size_t)(k0 + 2 * kh + 0) * N + ncol];
            b.y = W[(size_t)(k0 + 2 * kh + 1) * N + ncol];
            acc[t] = __builtin_amdgcn_wmma_f32_16x16x4_f32(
                false, a, false, b, (short)0, acc[t], false, false);
        }
    }

#pragma unroll
    for (int t = 0; t < NT; ++t) {
        const int ncol = (t << 4) + mr;
#pragma unroll
        for (int v = 0; v < 8; ++v) {
            const int row = m0 + 8 * kh + v;
            C[(size_t)row * N + ncol] = acc[t][v];
        }
    }
}

// ---------------------------------------------------------------------------
// edge scatter: agg[dst] += dis[src]*dis[dst] * h[src]
// one wave32 per edge; lanes stride the H features; fp32 HW atomics into
// L2-resident agg (25.6 MB << 192 MB L2)
// ---------------------------------------------------------------------------
template <int H>
__global__ __launch_bounds__(256) void edge_agg_k(const float* __restrict__ h,
                                                  const int* __restrict__ src,
                                                  const int* __restrict__ dst,
                                                  const float* __restrict__ dis,
                                                  float* __restrict__ agg, int nE) {
    const int warp = (blockIdx.x * blockDim.x + threadIdx.x) >> 5;
    const int lane = threadIdx.x & 31;
    if (warp >= nE) return;
    const int s = src[warp];
    const int d = dst[warp];
    const float norm = dis[s] * dis[d];
    const float* hs = h + (size_t)s * H;
    float* ad = agg + (size_t)d * H;
#pragma unroll
    for (int f = lane; f < H; f += 32)
        unsafeAtomicAdd(&ad[f], norm * hs[f]);
}

// ---------------------------------------------------------------------------
// epilogue: out = relu(agg + dis^2 * h + b)
// ---------------------------------------------------------------------------
template <int H>
__global__ __launch_bounds__(256) void finish_k(const float* __restrict__ agg,
                                                const float* __restrict__ h,
                                                const float* __restrict__ dis,
                                                const float* __restrict__ b,
                                                float* __restrict__ out, int M) {
    const int i = blockIdx.x * blockDim.x + threadIdx.x;
    if (i >= M * H) return;
    const int node = i >> (H == 64 ? 6 : 5);
    const int f    = i & (H - 1);
    const float dv = dis[node];
    const float v  = agg[i] + dv * dv * h[i] + b[f];
    out[i] = v > 0.0f ? v : 0.0f;
}

// ---------------------------------------------------------------------------
// row softmax over 32 columns: one wave32 per node, shuffle reductions
// ---------------------------------------------------------------------------
__global__ __launch_bounds__(256) void softmax32_k(float* __restrict__ out, int M) {
    const int warp = (blockIdx.x * blockDim.x + threadIdx.x) >> 5;
    const int lane = threadIdx.x & 31;
    if (warp >= M) return;
    const size_t idx = (size_t)warp * 32 + lane;
    const float v = out[idx];
    float m = v;
#pragma unroll
    for (int off = 16; off > 0; off >>= 1) m = fmaxf(m, __shfl_xor(m, off, 32));
    const float e = __expf(v - m);
    float s = e;
#pragma unroll
    for (int off = 16; off > 0; off >>= 1) s += __shfl_xor(s, off, 32);
    out[idx] = e / s;
}

// ---------------------------------------------------------------------------
// launch
// ---------------------------------------------------------------------------
extern "C" void kernel_launch(void* const* d_in, const int* in_sizes, int n_in,
                              void* d_out, int out_size, void* d_ws, size_t ws_size,
                              hipStream_t stream) {
    const float* x    = (const float*)d_in[0];
    const int*   edge = (const int*)d_in[1];   // [2, E] : row0 = src, row1 = dst
    const float* W1   = (const float*)d_in[2];
    const float* b1   = (const float*)d_in[3];
    const float* W2   = (const float*)d_in[4];
    const float* b2   = (const float*)d_in[5];
    const float* W3   = (const float*)d_in[6];
    const float* b3   = (const float*)d_in[7];
    float* out = (float*)d_out;

    const int* src = edge;
    const int* dst = edge + NEDGES;

    // workspace layout (all fp32):
    //   dis  : NNODES
    //   hbuf : NNODES*64   (GEMM output of current layer)
    //   abuf : NNODES*64   (scatter accumulator)
    //   xbuf : NNODES*64   (ReLU activation / next-layer input)
    float* dis  = (float*)d_ws;
    float* hbuf = dis  + NNODES;
    float* abuf = hbuf + (size_t)NNODES * 64;
    float* xbuf = abuf + (size_t)NNODES * 64;

    const int TB = 256;
    auto blocks = [](long long n, int tb) { return (int)((n + tb - 1) / tb); };

    const int gemm_waves  = NNODES / 16;               // 6250 (M divisible by 16)
    const int gemm_blocks = blocks((long long)gemm_waves * 32, TB);
    const int edge_blocks = blocks((long long)NEDGES * 32, TB);

    // ---- normalization -----------------------------------------------------
    zero_f32_k<<<blocks(NNODES, TB), TB, 0, stream>>>(dis, NNODES);
    degree_k<<<blocks(NEDGES, TB), TB, 0, stream>>>(dst, dis, NEDGES);
    rsqrt_k<<<blocks(NNODES, TB), TB, 0, stream>>>(dis, NNODES);

    // ---- layer 1: 128 -> 64 ------------------------------------------------
    gemm_wmma_f32_k<64><<<gemm_blocks, TB, 0, stream>>>(x, W1, hbuf, NNODES, INDIM);
    zero_f32_k<<<blocks((long long)NNODES * 64, TB), TB, 0, stream>>>(abuf, NNODES * 64);
    edge_agg_k<64><<<edge_blocks, TB, 0, stream>>>(hbuf, src, dst, dis, abuf, NEDGES);
    finish_k<64><<<blocks((long long)NNODES * 64, TB), TB, 0, stream>>>(abuf, hbuf, dis, b1, xbuf, NNODES);

    // ---- layer 2: 64 -> 64 -------------------------------------------------
    gemm_wmma_f32_k<64><<<gemm_blocks, TB, 0, stream>>>(xbuf, W2, hbuf, NNODES, HID1);
    zero_f32_k<<<blocks((long long)NNODES * 64, TB), TB, 0, stream>>>(abuf, NNODES * 64);
    edge_agg_k<64><<<edge_blocks, TB, 0, stream>>>(hbuf, src, dst, dis, abuf, NEDGES);
    finish_k<64><<<blocks((long long)NNODES * 64, TB), TB, 0, stream>>>(abuf, hbuf, dis, b2, xbuf, NNODES);

    // ---- layer 3: 64 -> 32 -------------------------------------------------
    gemm_wmma_f32_k<32><<<gemm_blocks, TB, 0, stream>>>(xbuf, W3, hbuf, NNODES, HID2);
    zero_f32_k<<<blocks((long long)NNODES * 32, TB), TB, 0, stream>>>(abuf, NNODES * 32);
    edge_agg_k<32><<<edge_blocks, TB, 0, stream>>>(hbuf, src, dst, dis, abuf, NEDGES);
    finish_k<32><<<blocks((long long)NNODES * 32, TB), TB, 0, stream>>>(abuf, hbuf, dis, b3, out, NNODES);

    // ---- softmax over 32 cols ---------------------------------------------
    softmax32_k<<<blocks((long long)NNODES * 32, TB), TB, 0, stream>>>(out, NNODES);
}